// SPP_21612275434379
// MI455X (gfx1250) — compile-verified
//
#include <hip/hip_runtime.h>

// SPP / SPPF: out = concat(x, maxpool5(x), maxpool9(x), maxpool13(x)) along C.
// Uses pool9 = pool5(pool5), pool13 = pool5(pool5(pool5)); each pool5 separable.
// One workgroup per 64x64 plane; plane lives in LDS, moved with gfx1250
// async LDS<->global instructions (ASYNCcnt) when available.

#define TPB    256
#define HH     64
#define WW     64
#define STRIDE 68            // 64 interior cols + 4 pad cols (shared left/right pad)
#define ROWS   68            // 2 pad rows top + 64 + 2 pad rows bottom
#define LDSZ   (ROWS * STRIDE)

#if defined(__HIP_DEVICE_COMPILE__) && defined(__gfx1250__) &&                 \
    __has_builtin(__builtin_amdgcn_global_load_async_to_lds_b128) &&           \
    __has_builtin(__builtin_amdgcn_global_store_async_from_lds_b128) &&        \
    __has_builtin(__builtin_amdgcn_s_wait_asynccnt)
#define HAS_ASYNC 1
#else
#define HAS_ASYNC 0
#endif

typedef int v4i __attribute__((ext_vector_type(4)));
typedef __attribute__((address_space(1))) v4i GInt4;  // global int4
typedef __attribute__((address_space(3))) v4i LInt4;  // LDS int4

__device__ __forceinline__ float max5(float a0, float a1, float a2, float a3, float a4) {
  return fmaxf(fmaxf(fmaxf(a0, a1), fmaxf(a2, a3)), a4);
}

__global__ __launch_bounds__(TPB) void spp_21612275434379(
    const float* __restrict__ x, float* __restrict__ out) {
  __shared__ float A[LDSZ];
  __shared__ float B[LDSZ];

  const int tid = threadIdx.x;
  const int p   = blockIdx.x;        // plane id = n*512 + c
  const int n   = p >> 9;
  const int c   = p & 511;
  const long long in_base  = (long long)p * (HH * WW);
  const long long out_base = ((long long)n * 2048 + c) * (HH * WW);
  const long long blkoff   = 512LL * HH * WW;

  const float NINF = -__builtin_inff();

  // Fill both buffers with -inf; compute only ever writes the interior, so the
  // halo (rows 0,1,66,67 and cols 64..67 of every row) stays -inf permanently.
  for (int i = tid; i < LDSZ; i += TPB) {
    A[i] = NINF;
    B[i] = NINF;
  }
  __syncthreads();

  // ---- DMA the 64x64 plane into A's interior (16 KB, 4 x b128 per thread) ----
#pragma unroll
  for (int i = 0; i < 4; ++i) {
    const int chunk = i * TPB + tid;        // 0..1023, 16 bytes each
    const int h     = chunk >> 4;
    const int s4    = (chunk & 15) * 4;
    const float* g  = x + in_base + 4LL * chunk;
    float*       l  = &A[(h + 2) * STRIDE + s4];
#if HAS_ASYNC
    __builtin_amdgcn_global_load_async_to_lds_b128((GInt4*)g, (LInt4*)l, 0, 0);
#else
    *(float4*)l = *(const float4*)g;
#endif
  }
#if HAS_ASYNC
  __builtin_amdgcn_s_wait_asynccnt(0);
#endif
  __syncthreads();

  // ---- output block 0: passthrough copy of x, streamed from LDS ----
#pragma unroll
  for (int i = 0; i < 4; ++i) {
    const int chunk = i * TPB + tid;
    const int h     = chunk >> 4;
    const int s4    = (chunk & 15) * 4;
    float* g        = out + out_base + 4LL * chunk;
    float* l        = &A[(h + 2) * STRIDE + s4];
#if HAS_ASYNC
    __builtin_amdgcn_global_store_async_from_lds_b128((GInt4*)g, (LInt4*)l, 0, 0);
#else
    *(float4*)g = *(const float4*)l;
#endif
  }

  // ---- three cascaded separable 5x5 max pools: A -> A, emitting each stage ----
  for (int stage = 1; stage <= 3; ++stage) {
    // Horizontal 5-max: each thread owns one row (h) x 16 cols [w0, w0+16).
    {
      const int h  = tid & 63;
      const int w0 = (tid >> 6) * 16;
      const float* rp = &A[(h + 2) * STRIDE + w0 - 2];  // w0-2 lands in -inf pad
      float a[20];
#pragma unroll
      for (int j = 0; j < 20; ++j) a[j] = rp[j];
      float* wp = &B[(h + 2) * STRIDE + w0];
#pragma unroll
      for (int j = 0; j < 16; ++j)
        wp[j] = max5(a[j], a[j + 1], a[j + 2], a[j + 3], a[j + 4]);
    }
    __syncthreads();

#if HAS_ASYNC
    // A is about to be overwritten; drain the async store that is reading it.
    __builtin_amdgcn_s_wait_asynccnt(0);
    __syncthreads();
#endif

    // Vertical 5-max: each thread owns one col (w) x 16 rows [h0, h0+16).
    {
      const int w  = tid & 63;
      const int h0 = (tid >> 6) * 16;
      float b[20];
#pragma unroll
      for (int j = 0; j < 20; ++j) b[j] = B[(h0 + j) * STRIDE + w];
#pragma unroll
      for (int r = 0; r < 16; ++r)
        A[(h0 + r + 2) * STRIDE + w] =
            max5(b[r], b[r + 1], b[r + 2], b[r + 3], b[r + 4]);
    }
    __syncthreads();

    // Stream stage result (now in A) to output block `stage`.
    const long long ob = out_base + (long long)stage * blkoff;
#pragma unroll
    for (int i = 0; i < 4; ++i) {
      const int chunk = i * TPB + tid;
      const int h     = chunk >> 4;
      const int s4    = (chunk & 15) * 4;
      float* g        = out + ob + 4LL * chunk;
      float* l        = &A[(h + 2) * STRIDE + s4];
#if HAS_ASYNC
      __builtin_amdgcn_global_store_async_from_lds_b128((GInt4*)g, (LInt4*)l, 0, 0);
#else
      *(float4*)g = *(const float4*)l;
#endif
    }
  }
  // Outstanding async stores are drained by S_ENDPGM's implicit wait-idle.
}

extern "C" void kernel_launch(void* const* d_in, const int* in_sizes, int n_in,
                              void* d_out, int out_size, void* d_ws, size_t ws_size,
                              hipStream_t stream) {
  (void)n_in; (void)out_size; (void)d_ws; (void)ws_size;
  const float* x = (const float*)d_in[0];
  float* out     = (float*)d_out;
  const int planes = in_sizes[0] >> 12;  // elements / (64*64) = 16*512 = 8192
  spp_21612275434379<<<dim3(planes), dim3(TPB), 0, stream>>>(x, out);
}